// MultiHeadAttention_58085137711126
// MI455X (gfx1250) — compile-verified
//
#include <hip/hip_runtime.h>

#define SEQ   1024
#define BATCH 8
#define HID   1024
#define NHEAD 16
#define DK    64                 // HID / NHEAD
#define HB    (NHEAD * BATCH)    // 128 head-batches
#define NROW  (SEQ * BATCH)      // 8192 rows for projection GEMMs

typedef __attribute__((ext_vector_type(2))) float v2f;
typedef __attribute__((ext_vector_type(8))) float v8f;

// D = A(16x4, f32) * B(4x16, f32) + C(16x16, f32), full-precision CDNA5 WMMA.
static __device__ __forceinline__ v8f wmma_16x16x4(v2f a, v2f b, v8f c) {
  return __builtin_amdgcn_wmma_f32_16x16x4_f32(false, a, false, b, (short)0, c,
                                               false, false);
}

// ---------------------------------------------------------------------------
// Kernel 1: P = X @ W^T with head-split store.
// X: (NROW, HID) row-major (row r = s*BATCH + b). W: (HID, HID) row-major.
// P[(s*HB + head*BATCH + b)*DK + d], where output col O = head*DK + d.
// Block tile 128x128, K-step 16, 8 waves each own a 16x128 strip.
// ---------------------------------------------------------------------------
__global__ __launch_bounds__(256) void mha_proj_split_kernel(
    const float* __restrict__ X, const float* __restrict__ W,
    float* __restrict__ P) {
  __shared__ float As[16][129];  // As[k][m] = X[rowBase+m][kb+k]
  __shared__ float Bs[16][129];  // Bs[k][n] = W[colBase+n][kb+k]  (= (W^T)[k][n])

  const int tid = threadIdx.x;
  const int wid = tid >> 5;
  const int lane = tid & 31;
  const int half = lane >> 4;   // K-pair / row-half select (wave32 layout)
  const int lc = lane & 15;     // M or N within 16
  const int colBase = blockIdx.x * 128;
  const int rowBase = blockIdx.y * 128;
  const int m0 = wid * 16;

  const v8f vzero = {0.f, 0.f, 0.f, 0.f, 0.f, 0.f, 0.f, 0.f};
  v8f acc[8];
#pragma unroll
  for (int j = 0; j < 8; ++j) acc[j] = vzero;

  for (int kb = 0; kb < HID; kb += 16) {
#pragma unroll
    for (int it = 0; it < 2; ++it) {
      const int idx = tid + it * 256;  // 0..511 float4 slots per matrix
      const int m = idx >> 2;          // 0..127
      const int k4 = (idx & 3) << 2;   // 0,4,8,12
      const float4 va = *reinterpret_cast<const float4*>(
          X + (size_t)(rowBase + m) * HID + kb + k4);
      As[k4 + 0][m] = va.x; As[k4 + 1][m] = va.y;
      As[k4 + 2][m] = va.z; As[k4 + 3][m] = va.w;
      const float4 vb = *reinterpret_cast<const float4*>(
          W + (size_t)(colBase + m) * HID + kb + k4);
      Bs[k4 + 0][m] = vb.x; Bs[k4 + 1][m] = vb.y;
      Bs[k4 + 2][m] = vb.z; Bs[k4 + 3][m] = vb.w;
    }
    if (kb + 16 < HID) {  // hint next A tile -> global_prefetch_b8
      __builtin_prefetch(X + (size_t)(rowBase + (tid >> 1)) * HID + kb + 16, 0, 1);
    }
    __syncthreads();
#pragma unroll
    for (int kk = 0; kk < 16; kk += 4) {
      v2f a;
      a[0] = As[kk + 2 * half + 0][m0 + lc];
      a[1] = As[kk + 2 * half + 1][m0 + lc];
#pragma unroll
      for (int j = 0; j < 8; ++j) {
        v2f b;
        b[0] = Bs[kk + 2 * half + 0][j * 16 + lc];
        b[1] = Bs[kk + 2 * half + 1][j * 16 + lc];
        acc[j] = wmma_16x16x4(a, b, acc[j]);
      }
    }
    __syncthreads();
  }

#pragma unroll
  for (int j = 0; j < 8; ++j) {
    const int O = colBase + j * 16 + lc;
    const int head = O >> 6;
    const int d = O & 63;
#pragma unroll
    for (int r = 0; r < 8; ++r) {
      const int R = rowBase + m0 + r + 8 * half;  // C/D layout: row = r + 8*(lane/16)
      const int s = R >> 3;
      const int b = R & 7;
      P[((size_t)(s * HB + head * BATCH + b)) * DK + d] = acc[j][r];
    }
  }
}

// ---------------------------------------------------------------------------
// Kernel 2: fused scores + causal mask + softmax(axis=j) -> dist.
// For head-batch hb: scores[i][j] = (K_i . Q_j) * 0.125, masked j>i,
// softmax over j. Block owns 16 rows x 1024 cols; wave w owns cols
// [w*128, w*128+128). Softmax reductions done in registers + shfl_xor
// within 16-lane halves + small LDS across waves. dist written once.
// ---------------------------------------------------------------------------
__global__ __launch_bounds__(256) void mha_scores_softmax_kernel(
    const float* __restrict__ Kp, const float* __restrict__ Qp,
    float* __restrict__ dist) {
  __shared__ float Ks[64][17];   // Ks[k][m] = K[iBase+m][k]
  __shared__ float Rmax[8][16];  // [wave][local row]
  __shared__ float Rsum[8][16];

  const int tid = threadIdx.x;
  const int wid = tid >> 5;
  const int lane = tid & 31;
  const int half = lane >> 4;
  const int lc = lane & 15;
  const int iBase = blockIdx.x * 16;
  const int hb = blockIdx.y;
  const int colBase = wid * 128;

  {  // stage transposed K tile (16 rows x 64 d)
    const int m = tid >> 4;          // 0..15
    const int d4 = (tid & 15) << 2;  // 0..60
    const float4 v = *reinterpret_cast<const float4*>(
        Kp + ((size_t)(iBase + m) * HB + hb) * DK + d4);
    Ks[d4 + 0][m] = v.x; Ks[d4 + 1][m] = v.y;
    Ks[d4 + 2][m] = v.z; Ks[d4 + 3][m] = v.w;
  }
  __syncthreads();

  const v8f vzero = {0.f, 0.f, 0.f, 0.f, 0.f, 0.f, 0.f, 0.f};
  v8f acc[8];
#pragma unroll
  for (int j = 0; j < 8; ++j) acc[j] = vzero;

#pragma unroll
  for (int kk = 0; kk < DK; kk += 4) {
    v2f a;
    a[0] = Ks[kk + 2 * half + 0][lc];
    a[1] = Ks[kk + 2 * half + 1][lc];
#pragma unroll
    for (int j = 0; j < 8; ++j) {
      const int col = colBase + j * 16 + lc;
      // B[k][j] = Q[col][k]; contiguous float2, 8B aligned (k-pair even)
      const float2 q2 = *reinterpret_cast<const float2*>(
          Qp + ((size_t)col * HB + hb) * DK + kk + 2 * half);
      v2f b; b[0] = q2.x; b[1] = q2.y;
      acc[j] = wmma_16x16x4(a, b, acc[j]);
    }
  }

  // scale + causal mask (mask where query index j > key index i)
#pragma unroll
  for (int j = 0; j < 8; ++j) {
    const int col = colBase + j * 16 + lc;
#pragma unroll
    for (int r = 0; r < 8; ++r) {
      const int row = iBase + 8 * half + r;
      float v = acc[j][r] * 0.125f;  // 1/sqrt(64)
      if (col > row) v = -__builtin_inff();
      acc[j][r] = v;
    }
  }

  // row max: per-lane over 8 accs, shfl_xor over the 16-lane half, LDS across waves
  float pm[8];
#pragma unroll
  for (int r = 0; r < 8; ++r) {
    float m = acc[0][r];
#pragma unroll
    for (int j = 1; j < 8; ++j) m = fmaxf(m, acc[j][r]);
#pragma unroll
    for (int off = 1; off < 16; off <<= 1) m = fmaxf(m, __shfl_xor(m, off, 32));
    pm[r] = m;
  }
  if (lc == 0) {
#pragma unroll
    for (int r = 0; r < 8; ++r) Rmax[wid][8 * half + r] = pm[r];
  }
  __syncthreads();
  float gm[8];
#pragma unroll
  for (int r = 0; r < 8; ++r) {
    float m = Rmax[0][8 * half + r];
#pragma unroll
    for (int w = 1; w < 8; ++w) m = fmaxf(m, Rmax[w][8 * half + r]);
    gm[r] = m;
  }

  // exp + row sum  (exp(-inf) == 0 handles the mask)
  float ps[8];
#pragma unroll
  for (int r = 0; r < 8; ++r) ps[r] = 0.f;
#pragma unroll
  for (int j = 0; j < 8; ++j) {
#pragma unroll
    for (int r = 0; r < 8; ++r) {
      const float e = __expf(acc[j][r] - gm[r]);
      acc[j][r] = e;
      ps[r] += e;
    }
  }
#pragma unroll
  for (int r = 0; r < 8; ++r) {
    float s = ps[r];
#pragma unroll
    for (int off = 1; off < 16; off <<= 1) s += __shfl_xor(s, off, 32);
    ps[r] = s;
  }
  if (lc == 0) {
#pragma unroll
    for (int r = 0; r < 8; ++r) Rsum[wid][8 * half + r] = ps[r];
  }
  __syncthreads();
  float inv[8];
#pragma unroll
  for (int r = 0; r < 8; ++r) {
    float s = Rsum[0][8 * half + r];
#pragma unroll
    for (int w = 1; w < 8; ++w) s += Rsum[w][8 * half + r];
    inv[r] = 1.0f / s;
  }

  const size_t dbase = (size_t)hb * SEQ * SEQ;
#pragma unroll
  for (int j = 0; j < 8; ++j) {
    const int col = colBase + j * 16 + lc;
#pragma unroll
    for (int r = 0; r < 8; ++r) {
      const int row = iBase + 8 * half + r;
      dist[dbase + (size_t)row * SEQ + col] = acc[j][r] * inv[r];
    }
  }
}

// ---------------------------------------------------------------------------
// Kernel 3: att_b = dist_b^T @ V_b  -> Att[(j*HB + hb)*DK + d]
// Block: 128 j-rows x 64 d-cols per head-batch; K (= i) swept in 32-chunks.
// ---------------------------------------------------------------------------
__global__ __launch_bounds__(256) void mha_att_kernel(
    const float* __restrict__ dist, const float* __restrict__ Vp,
    float* __restrict__ Att) {
  __shared__ float Ds[32][129];  // Ds[k][m] = dist[ib+k][jBase+m]  (A = dist^T)
  __shared__ float Vs[32][65];   // Vs[k][n] = V[ib+k][n]

  const int tid = threadIdx.x;
  const int wid = tid >> 5;
  const int lane = tid & 31;
  const int half = lane >> 4;
  const int lc = lane & 15;
  const int jBase = blockIdx.x * 128;
  const int hb = blockIdx.y;
  const int m0 = wid * 16;
  const size_t dbase = (size_t)hb * SEQ * SEQ;

  const v8f vzero = {0.f, 0.f, 0.f, 0.f, 0.f, 0.f, 0.f, 0.f};
  v8f acc[4];
#pragma unroll
  for (int j = 0; j < 4; ++j) acc[j] = vzero;

  for (int ib = 0; ib < SEQ; ib += 32) {
#pragma unroll
    for (int it = 0; it < 4; ++it) {  // 32x128 dist tile, coalesced rows
      const int idx = tid + it * 256;
      const int k = idx >> 5;
      const int m4 = (idx & 31) << 2;
      const float4 v = *reinterpret_cast<const float4*>(
          dist + dbase + (size_t)(ib + k) * SEQ + jBase + m4);
      Ds[k][m4 + 0] = v.x; Ds[k][m4 + 1] = v.y;
      Ds[k][m4 + 2] = v.z; Ds[k][m4 + 3] = v.w;
    }
#pragma unroll
    for (int it = 0; it < 2; ++it) {  // 32x64 V tile
      const int idx = tid + it * 256;
      const int k = idx >> 4;
      const int n4 = (idx & 15) << 2;
      const float4 v = *reinterpret_cast<const float4*>(
          Vp + ((size_t)(ib + k) * HB + hb) * DK + n4);
      Vs[k][n4 + 0] = v.x; Vs[k][n4 + 1] = v.y;
      Vs[k][n4 + 2] = v.z; Vs[k][n4 + 3] = v.w;
    }
    __syncthreads();
#pragma unroll
    for (int kk = 0; kk < 32; kk += 4) {
      v2f a;
      a[0] = Ds[kk + 2 * half + 0][m0 + lc];
      a[1] = Ds[kk + 2 * half + 1][m0 + lc];
#pragma unroll
      for (int j = 0; j < 4; ++j) {
        v2f b;
        b[0] = Vs[kk + 2 * half + 0][j * 16 + lc];
        b[1] = Vs[kk + 2 * half + 1][j * 16 + lc];
        acc[j] = wmma_16x16x4(a, b, acc[j]);
      }
    }
    __syncthreads();
  }

#pragma unroll
  for (int j = 0; j < 4; ++j) {
    const int d = j * 16 + lc;
#pragma unroll
    for (int r = 0; r < 8; ++r) {
      const int J = jBase + m0 + 8 * half + r;
      Att[((size_t)J * HB + hb) * DK + d] = acc[j][r];
    }
  }
}

// ---------------------------------------------------------------------------
// Kernel 4: out = att_r @ Wo^T, gathering A from the head-split Att buffer.
// Row R = s*BATCH + b, so the store R*HID + O is exactly (S,B,H) layout.
// A[R][h] = Att[(s*HB + (h>>6)*BATCH + b)*DK + (h&63)].
// ---------------------------------------------------------------------------
__global__ __launch_bounds__(256) void mha_outproj_kernel(
    const float* __restrict__ Att, const float* __restrict__ Wo,
    float* __restrict__ out) {
  __shared__ float As[16][129];
  __shared__ float Bs[16][129];

  const int tid = threadIdx.x;
  const int wid = tid >> 5;
  const int lane = tid & 31;
  const int half = lane >> 4;
  const int lc = lane & 15;
  const int colBase = blockIdx.x * 128;
  const int rowBase = blockIdx.y * 128;
  const int m0 = wid * 16;

  const v8f vzero = {0.f, 0.f, 0.f, 0.f, 0.f, 0.f, 0.f, 0.f};
  v8f acc[8];
#pragma unroll
  for (int j = 0; j < 8; ++j) acc[j] = vzero;

  for (int kb = 0; kb < HID; kb += 16) {
    const int head = kb >> 6;    // 16-wide K tile stays inside one head block
    const int dBase = kb & 63;
#pragma unroll
    for (int it = 0; it < 2; ++it) {
      const int idx = tid + it * 256;
      const int m = idx >> 2;
      const int k4 = (idx & 3) << 2;
      const int R = rowBase + m;
      const int s = R >> 3;
      const int b = R & 7;
      const float4 va = *reinterpret_cast<const float4*>(
          Att + ((size_t)(s * HB + head * BATCH + b)) * DK + dBase + k4);
      As[k4 + 0][m] = va.x; As[k4 + 1][m] = va.y;
      As[k4 + 2][m] = va.z; As[k4 + 3][m] = va.w;
      const float4 vb = *reinterpret_cast<const float4*>(
          Wo + (size_t)(colBase + m) * HID + kb + k4);
      Bs[k4 + 0][m] = vb.x; Bs[k4 + 1][m] = vb.y;
      Bs[k4 + 2][m] = vb.z; Bs[k4 + 3][m] = vb.w;
    }
    __syncthreads();
#pragma unroll
    for (int kk = 0; kk < 16; kk += 4) {
      v2f a;
      a[0] = As[kk + 2 * half + 0][m0 + lc];
      a[1] = As[kk + 2 * half + 1][m0 + lc];
#pragma unroll
      for (int j = 0; j < 8; ++j) {
        v2f b;
        b[0] = Bs[kk + 2 * half + 0][j * 16 + lc];
        b[1] = Bs[kk + 2 * half + 1][j * 16 + lc];
        acc[j] = wmma_16x16x4(a, b, acc[j]);
      }
    }
    __syncthreads();
  }

#pragma unroll
  for (int j = 0; j < 8; ++j) {
    const int O = colBase + j * 16 + lc;
#pragma unroll
    for (int r = 0; r < 8; ++r) {
      const int R = rowBase + m0 + r + 8 * half;
      out[(size_t)R * HID + O] = acc[j][r];
    }
  }
}

// ---------------------------------------------------------------------------
extern "C" void kernel_launch(void* const* d_in, const int* in_sizes, int n_in,
                              void* d_out, int out_size, void* d_ws,
                              size_t ws_size, hipStream_t stream) {
  (void)in_sizes; (void)n_in; (void)out_size; (void)ws_size;
  const float* value = (const float*)d_in[0];
  const float* key   = (const float*)d_in[1];
  const float* query = (const float*)d_in[2];
  const float* Wq    = (const float*)d_in[3];
  const float* Wk    = (const float*)d_in[4];
  const float* Wv    = (const float*)d_in[5];
  const float* Wo    = (const float*)d_in[6];

  float* out  = (float*)d_out;                         // (S,B,H)
  float* dist = out + (size_t)SEQ * BATCH * HID;       // (HB,S,S), 2nd output

  const size_t PROJ = (size_t)SEQ * HB * DK;           // 8,388,608 floats
  float* ws = (float*)d_ws;                            // 128 MiB used
  float* Qp = ws;
  float* Kp = ws + PROJ;
  float* Vp = ws + 2 * PROJ;
  float* Ap = ws + 3 * PROJ;

  const dim3 blk(256);
  const dim3 gproj(HID / 128, NROW / 128);  // (8, 64)

  mha_proj_split_kernel<<<gproj, blk, 0, stream>>>(query, Wq, Qp);
  mha_proj_split_kernel<<<gproj, blk, 0, stream>>>(key,   Wk, Kp);
  mha_proj_split_kernel<<<gproj, blk, 0, stream>>>(value, Wv, Vp);
  mha_scores_softmax_kernel<<<dim3(SEQ / 16, HB), blk, 0, stream>>>(Kp, Qp, dist);
  mha_att_kernel<<<dim3(SEQ / 128, HB), blk, 0, stream>>>(dist, Vp, Ap);
  mha_outproj_kernel<<<gproj, blk, 0, stream>>>(Ap, Wo, out);
}